// lstm_84567906058356
// MI455X (gfx1250) — compile-verified
//
#include <hip/hip_runtime.h>

// D = A(16x4) * B(4x16) + C(16x16), f32, one wave (wave32), EXEC all-ones.
typedef __attribute__((ext_vector_type(2))) float v2f;
typedef __attribute__((ext_vector_type(8))) float v8f;

#define LMAX 4096                      // reference L = 2048
#define L2E  1.4426950408889634f       // log2(e)

__device__ __forceinline__ float bperm(int byte_idx, float v) {
    return __int_as_float(__builtin_amdgcn_ds_bpermute(byte_idx, __float_as_int(v)));
}

// Single-wave sequential LSTM over the ONE batch element the output depends on
// (out = hs[:, B-1, :] @ w_fc.T + b_fc  ->  batch elements 0..B-2 are dead).
//
// Per step, ONE v_wmma_f32_16x16x4_f32 computes BOTH:
//   lanes 0-11 : gates = [h,x_t](1x4) x [w_hh^T ; w_ih^T](4x12) + (b_ih+b_hh)
//   lane  12   : out[t-1] = h_{t-1} . w_fc + b_fc   (free FC projection)
__global__ __launch_bounds__(32)
void lstm_seq_wmma_kernel(const float* __restrict__ x,
                          const float* __restrict__ w_ih,   // (12,1)
                          const float* __restrict__ w_hh,   // (12,3) row-major
                          const float* __restrict__ b_ih,   // (12)
                          const float* __restrict__ b_hh,   // (12)
                          const float* __restrict__ w_fc,   // (1,3)
                          const float* __restrict__ b_fc,   // (1)
                          float* __restrict__ out,          // (L)
                          int L, int B)
{
    __shared__ float xs[LMAX];
    const int lane = threadIdx.x;

    // Stage x[:, B-1, 0] into LDS once (2048 independent strided loads).
    #pragma unroll 4
    for (int t = lane; t < L; t += 32)
        xs[t] = x[(size_t)t * (size_t)B + (size_t)(B - 1)];
    __syncthreads();

    // ---- Constant B matrix (4x16) ----
    // cols 0-11: rows K0..K2 = w_hh^T, row K3 = w_ih^T   (gate weights)
    // col  12  : rows K0..K2 = w_fc,   row K3 = 0        (FC projection)
    // 32-bit B layout: lane = N (0-15 -> K0/K1 in VGPR0/1; 16-31 -> K2/K3).
    const int n  = lane & 15;
    const int hi = lane >> 4;
    v2f bm; bm[0] = 0.0f; bm[1] = 0.0f;
    if (n < 12) {
        bm[0] = hi ? w_hh[n * 3 + 2] : w_hh[n * 3 + 0];  // K2 : K0
        bm[1] = hi ? w_ih[n]         : w_hh[n * 3 + 1];  // K3 : K1
    } else if (n == 12) {
        bm[0] = hi ? w_fc[2] : w_fc[0];                  // K2 : K0
        bm[1] = hi ? 0.0f    : w_fc[1];                  // K3 : K1
    }

    // ---- Constant C: row M=0 = gate bias (lanes 0-11) / b_fc (lane 12) ----
    v8f cm = {};
    if (hi == 0) {
        if (n < 12)       cm[0] = b_ih[n] + b_hh[n];
        else if (n == 12) cm[0] = b_fc[0];
    }

    // Per-lane activation constants: lanes 0-5,9-11 sigmoid; 6-8 tanh.
    //   sigmoid(x) = rcp(1 + 2^(-x*L2E))
    //   tanh(x)    = 1 - 2*rcp(1 + 2^(2x*L2E))
    const bool  g_is_tanh = (lane >= 6 && lane < 9);
    const float escale = g_is_tanh ?  2.0f * L2E : -L2E;
    const float asc    = g_is_tanh ? -2.0f : 1.0f;
    const float aoff   = g_is_tanh ?  1.0f : 0.0f;

    // Precomputed bpermute BYTE indices (loop-invariant; no per-iter `% 3`).
    const int base = lane % 3;               // hidden unit this lane tracks
    const int ixf  = (base + 3) * 4;         // f-gate source lane
    const int ixg  = (base + 6) * 4;         // g-gate source lane
    const int ixo  = (base + 9) * 4;         // o-gate source lane
    // Next-A operands: lane16 needs h2 (lane 2), lane0 needs own h0 / h1 (lane1).
    const int ixa0 = (lane == 16) ? (2 * 4) : (lane * 4);
    const int ixa1 = 1 * 4;

    const bool is0   = (lane == 0);
    const bool is16  = (lane == 16);
    const bool is016 = is0 | is16;

    const float wf2 = w_fc[2], bfc = b_fc[0];   // epilogue only
    const float wf0 = w_fc[0], wf1 = w_fc[1];

    float c   = 0.0f;      // cell state for unit `base` (lanes 0-2 authoritative)
    float pa0 = 0.0f;      // next A[0]: h0 on lane0 / h2 on lane16
    float pa1 = 0.0f;      // next A[1]: h1 (on lane0)
    float h   = 0.0f;      // this lane's hidden unit value
    float xt  = xs[0];

    for (int t = 0; t < L; ++t) {
        // Prefetch next x early: its dscnt wait overlaps the whole body.
        const float xt_n = xs[(t + 1 < L) ? t + 1 : 0];

        // A row 0 = [h0, h1, h2, x_t]; rows 1..15 zero.
        // Layout: VGPR0 lane0=K0, lane16=K2 ; VGPR1 lane0=K1, lane16=K3.
        v2f a;
        a[0] = is016 ? pa0 : 0.0f;
        a[1] = is0 ? pa1 : (is16 ? xt : 0.0f);

        v8f d = __builtin_amdgcn_wmma_f32_16x16x4_f32(
            /*neg_a=*/false, a, /*neg_b=*/false, bm,
            /*c_mod=*/(short)0, cm, /*reuse_a=*/false, /*reuse_b=*/false);

        const float gd = d[0];              // row 0: gates (0-11) / out[t-1] (12)
        if (lane == 12 && t)                // fused FC output of previous step
            out[t - 1] = gd;

        // Activation: v_mul -> v_exp_f32 -> v_add -> v_rcp_f32 -> v_fma.
        const float e   = __builtin_amdgcn_exp2f(gd * escale);
        const float r   = __builtin_amdgcn_rcpf(1.0f + e);
        const float act = __builtin_fmaf(asc, r, aoff);

        // ai is LOCAL on lanes 0-2 (they are the i-gate lanes); gather f/g/o.
        const float af = bperm(ixf, act);
        const float ag = bperm(ixg, act);
        const float ao = bperm(ixo, act);

        c = __builtin_fmaf(af, c, act * ag);                   // cell update
        const float e2 = __builtin_amdgcn_exp2f(c * (2.0f * L2E));
        const float tc = __builtin_fmaf(-2.0f, __builtin_amdgcn_rcpf(1.0f + e2), 1.0f);
        h = ao * tc;                                           // new hidden

        // Route h into next step's A operands (2 bpermutes, not 3 broadcasts).
        pa0 = bperm(ixa0, h);   // lane0 <- h0 (self), lane16 <- h2 (lane 2)
        pa1 = bperm(ixa1, h);   // all lanes <- h1 (only lane0's copy is used)
        xt  = xt_n;
    }

    // Final timestep's projection was never emitted by the WMMA pipeline.
    const float h2f = __shfl(h, 2);
    if (lane == 0)
        out[L - 1] = pa0 * wf0 + pa1 * wf1 + h2f * wf2 + bfc;  // pa0=h0, pa1=h1
}

extern "C" void kernel_launch(void* const* d_in, const int* in_sizes, int n_in,
                              void* d_out, int out_size, void* d_ws, size_t ws_size,
                              hipStream_t stream) {
    const float* x    = (const float*)d_in[0];
    const float* w_ih = (const float*)d_in[1];
    const float* w_hh = (const float*)d_in[2];
    const float* b_ih = (const float*)d_in[3];
    const float* b_hh = (const float*)d_in[4];
    const float* w_fc = (const float*)d_in[5];
    const float* b_fc = (const float*)d_in[6];
    float* out = (float*)d_out;

    const int L = out_size;                 // 2048: one output per timestep
    const int B = in_sizes[0] / L;          // 8192: only batch B-1 is live

    lstm_seq_wmma_kernel<<<1, 32, 0, stream>>>(x, w_ih, w_hh, b_ih, b_hh,
                                               w_fc, b_fc, out, L, B);
}